// PointCloudSubsampling_10393820856387
// MI455X (gfx1250) — compile-verified
//
#include <hip/hip_runtime.h>
#include <cstdint>

#define NBATCH 16
#define NPTS   131072
#define NCH    6
#define KSEL   1024
#define TPB    1024
#define NWAVE  32              // TPB / 32 (wave32)
#define PW     (NPTS / NWAVE)  // 4096 points per wave
#define SUB    128             // points per staged subtile
#define NSUB   (PW / SUB)      // 32 subtiles per wave per sweep

typedef unsigned int u32x4 __attribute__((ext_vector_type(4)));
typedef int          i32x8 __attribute__((ext_vector_type(8)));
typedef int          i32x4 __attribute__((ext_vector_type(4)));

#if defined(__has_builtin)
#if __has_builtin(__builtin_amdgcn_tensor_load_to_lds) && __has_builtin(__builtin_amdgcn_s_wait_tensorcnt)
#define USE_TDM 1
#endif
#endif

__global__ __launch_bounds__(TPB) void fps_kernel(const float* __restrict__ pts,
                                                  float* __restrict__ out,
                                                  float* __restrict__ dists)
{
    // per-wave double-buffered xyz staging: 32 waves * 2 * 128 pts * 3 ch * 4B = 96 KB
    __shared__ float stage[NWAVE][2][SUB * 3];
    __shared__ float s_p[3];
    __shared__ float s_wv[NWAVE];
    __shared__ int   s_wi[NWAVE];
    __shared__ int   s_sel[KSEL];

    const int    b     = blockIdx.x;
    const int    tid   = threadIdx.x;
    const int    wave  = tid >> 5;
    const int    lane  = tid & 31;
    const size_t baseP = (size_t)b * NPTS;
    float* __restrict__ dmin = dists + baseP;

    // ---- init: start index 0, dists = +inf (matches jnp.full(inf) + min) ----
    if (tid == 0) {
        s_sel[0] = 0;
        s_p[0] = pts[baseP * NCH + 0];
        s_p[1] = pts[baseP * NCH + 1];
        s_p[2] = pts[baseP * NCH + 2];
    }
    {
        const int n0w = wave * PW;
        for (int s = 0; s < NSUB; ++s) {
            const int base = n0w + s * SUB + lane;
#pragma unroll
            for (int h = 0; h < SUB / 32; ++h)
                dmin[base + h * 32] = __builtin_inff();
        }
    }
    __syncthreads();

#ifdef USE_TDM
    // Wave-uniform (SGPR) tensor-DMA descriptor pieces.
    // Tile = 3 channels (xyz) x SUB points out of the [N,6] fp32 AoS array.
    const int uwave = __builtin_amdgcn_readfirstlane(wave);
    i32x8 g1;
    g1[0] = 0x20000;                      // data_size = 4B (code 2 at bits 17:16)
    g1[1] = (int)(6u << 16);              // tensor_dim0 = 6 (bits 63:48 = low16)
    g1[2] = (int)((unsigned)SUB << 16);   // dim0 hi16=0 | tensor_dim1 = SUB
    g1[3] = (int)(3u << 16);              // dim1 hi16=0 | tile_dim0 = 3 (xyz)
    g1[4] = SUB;                          // tile_dim1 = SUB, tile_dim2 = 0
    g1[5] = 6;                            // tensor_dim0_stride = 6 dwords/point
    g1[6] = 0;
    g1[7] = 0;
    const i32x4 gz4 = {0, 0, 0, 0};
    const i32x8 gz8 = {0, 0, 0, 0, 0, 0, 0, 0};
    const uint64_t gbase = (uint64_t)(uintptr_t)pts + (uint64_t)baseP * (NCH * 4u);
    unsigned lds0 = (unsigned)(uintptr_t)(&stage[uwave][0][0]);
    unsigned lds1 = (unsigned)(uintptr_t)(&stage[uwave][1][0]);
    lds0 = __builtin_amdgcn_readfirstlane((int)lds0);
    lds1 = __builtin_amdgcn_readfirstlane((int)lds1);
#endif

    // ---- K-1 sequential FPS iterations ----
    for (int k = 1; k < KSEL; ++k) {
        const float px = s_p[0], py = s_p[1], pz = s_p[2];
        float bestv = -1.0f;
        int   besti = 0;
        const int n0w = wave * PW;

#ifdef USE_TDM
        auto issue = [&](int s, unsigned ldsaddr) {
            const uint64_t ga = gbase + (uint64_t)(n0w + s * SUB) * (NCH * 4u);
            u32x4 g0;
            g0[0] = 1u;                                     // one valid descriptor
            g0[1] = ldsaddr;                                // LDS byte address
            g0[2] = (unsigned)__builtin_amdgcn_readfirstlane((int)(unsigned)(ga & 0xFFFFFFFFull));
            g0[3] = (unsigned)__builtin_amdgcn_readfirstlane(
                        (int)((unsigned)((ga >> 32) & 0x1FFFFFFull) | (2u << 30))); // type=2
            asm volatile("" ::: "memory");                  // pin LDS ordering vs DMA
            __builtin_amdgcn_tensor_load_to_lds(g0, g1, gz4, gz4, gz8, 0);
        };
        issue(0, lds0);
#endif
        for (int s = 0; s < NSUB; ++s) {
            const int n0 = n0w + s * SUB;
#ifdef USE_TDM
            if (s + 1 < NSUB) {
                issue(s + 1, ((s + 1) & 1) ? lds1 : lds0);
                __builtin_prefetch(&dmin[n0 + SUB + lane], 0, 3);
                __builtin_amdgcn_s_wait_tensorcnt(1);       // subtile s landed in LDS
            } else {
                __builtin_amdgcn_s_wait_tensorcnt(0);
            }
            const float* bp = &stage[wave][s & 1][0];
#endif
#pragma unroll
            for (int h = 0; h < SUB / 32; ++h) {
                const int p = lane + h * 32;
                const int n = n0 + p;
#ifdef USE_TDM
                const float x = bp[p * 3 + 0];
                const float y = bp[p * 3 + 1];
                const float z = bp[p * 3 + 2];
#else
                const float* q = pts + (baseP + (size_t)n) * NCH;
                const float x = q[0], y = q[1], z = q[2];
#endif
                const float dx = x - px, dy = y - py, dz = z - pz;
                const float d  = dx * dx + dy * dy + dz * dz;
                float dm = dmin[n];
                dm = fminf(dm, d);
                dmin[n] = dm;
                if (dm > bestv) { bestv = dm; besti = n; }   // ascending n -> first-max
            }
        }

        // wave32 argmax reduce (ties -> lowest index)
        for (int o = 16; o > 0; o >>= 1) {
            const float ov = __shfl_down(bestv, o, 32);
            const int   oi = __shfl_down(besti, o, 32);
            if (ov > bestv || (ov == bestv && oi < besti)) { bestv = ov; besti = oi; }
        }
        if (lane == 0) { s_wv[wave] = bestv; s_wi[wave] = besti; }
        __syncthreads();
        if (wave == 0) {
            float v = s_wv[lane];
            int   i = s_wi[lane];
            for (int o = 16; o > 0; o >>= 1) {
                const float ov = __shfl_down(v, o, 32);
                const int   oi = __shfl_down(i, o, 32);
                if (ov > v || (ov == v && oi < i)) { v = ov; i = oi; }
            }
            if (lane == 0) {
                s_sel[k] = i;
                const float* q = pts + (baseP + (size_t)i) * NCH;
                s_p[0] = q[0]; s_p[1] = q[1]; s_p[2] = q[2];
            }
        }
        __syncthreads();
    }

    // ---- gather: out[b,k,c] = pts[b, sel[k], c] ----
    for (int e = tid; e < KSEL * NCH; e += TPB) {
        const int k = e / NCH;
        const int c = e % NCH;
        out[((size_t)b * KSEL + k) * NCH + c] =
            pts[(baseP + (size_t)s_sel[k]) * NCH + c];
    }
}

extern "C" void kernel_launch(void* const* d_in, const int* in_sizes, int n_in,
                              void* d_out, int out_size, void* d_ws, size_t ws_size,
                              hipStream_t stream) {
    const float* pts = (const float*)d_in[0];   // [16,131072,6] fp32
    float* out   = (float*)d_out;               // [16,1024,6]   fp32
    float* dists = (float*)d_ws;                // needs 16*131072*4 = 8 MB scratch
    (void)in_sizes; (void)n_in; (void)out_size; (void)ws_size;
    fps_kernel<<<dim3(NBATCH), dim3(TPB), 0, stream>>>(pts, out, dists);
}